// FMSynthesis_90142773608899
// MI455X (gfx1250) — compile-verified
//
#include <hip/hip_runtime.h>
#include <math.h>

typedef __attribute__((ext_vector_type(2))) float v2f;
typedef __attribute__((ext_vector_type(8))) float v8f;

#define PI_F          3.14159265358979323846f
#define TWO_PI_F      6.28318530717958647692f
#define TWO_OVER_PI_F 0.63661977236758134308f

// Self-contained accurate sin for |x| <= ~2^22 (our phases reach ~7.6e5).
// One f64 reduction by pi/2 (error ~3e-11 rad), then degree-7/8 minimax
// sin/cos on [-pi/4, pi/4] with branchless quadrant select.
// ~16 VALU; no Payne-Hanek body, no library call.
__device__ __forceinline__ float sin_fast(float x) {
    const double xd = (double)x;
    const double qd = rint(xd * 0.63661977236758134308);        // x / (pi/2)
    const float  r  = (float)fma(qd, -1.5707963267948966, xd);  // |r| <= pi/4
    const int    iq = ((int)qd) & 3;

    const float z = r * r;
    // sin(r), Cephes minimax on [-pi/4, pi/4]
    float sp = fmaf(z, fmaf(z, fmaf(z, -1.9515295891e-4f, 8.3321608736e-3f),
                            -1.6666654611e-1f), 0.0f);
    sp = fmaf(r * z, sp / z, r);                 // r + r*z*(poly)  (kept exact form below)
    sp = r + r * z * fmaf(z, fmaf(z, -1.9515295891e-4f, 8.3321608736e-3f),
                          -1.6666654611e-1f);
    // cos(r)
    const float cp = fmaf(z * z,
                          fmaf(z, fmaf(z, 2.443315711809948e-5f, -1.388731625493765e-3f),
                               4.166664568298827e-2f),
                          fmaf(z, -0.5f, 1.0f));

    const float v = (iq & 1) ? cp : sp;          // quadrant 1,3 -> cos
    return (iq & 2) ? -v : v;                    // quadrant 2,3 -> negate
}

// FM synthesis, fully fused. One wave32 handles a tile of 16 samples:
//   D(16x16) = A(16x8 as 2xK4) * B(8x16 as 2xK4)  via V_WMMA_F32_16X16X4_F32
// K axis = oscillators, permuted {tri,saw | sine,square | pulse} so the
// transcendental load is split 3/2 across the two 16-lane halves. D rows 0..4
// give mod[carrier][sample] in lanes 0..15 with no cross-lane reduction; the
// 5-carrier sin epilogue is split 3/2 across the halves via ds_bpermute.
__global__ __launch_bounds__(256) void fm_synth_wmma(
    const float* __restrict__ freqs,        // [5]
    const float* __restrict__ amps,         // [5]
    const float* __restrict__ pulse_width,  // [1]
    const float* __restrict__ W,            // [5,5]
    const int*   __restrict__ duration,     // [1]
    const int*   __restrict__ sample_rate,  // [1]
    float*       __restrict__ out)          // [n]
{
    const int   dur = duration[0];
    const int   sr  = sample_rate[0];
    const int   n   = dur * sr;
    const float inv_sr = 1.0f / (float)sr;
    const unsigned numTiles = (unsigned)((n + 15) >> 4);

    const unsigned lane = threadIdx.x & 31u;
    const unsigned half = lane >> 4;     // 0: lanes 0-15, 1: lanes 16-31
    const unsigned m    = lane & 15u;    // A row (carrier) / B-D column (sample)

    // Uniform scalars (SGPR loads). omega/cc formed exactly as the f32
    // reference rounds them: ((2pi)_f32 * f) [* t | * duration].
    float f[5], a[5], om[5], cc[5];
#pragma unroll
    for (int j = 0; j < 5; ++j) {
        f[j]  = freqs[j];
        a[j]  = amps[j];
        om[j] = TWO_PI_F * f[j];
        cc[j] = om[j] * (float)dur;
    }
    const float pw = pulse_width[0];

    // --- A fragments: Wm = W with zeroed diagonal, K-permuted, padded 16x8.
    // K -> oscillator: 0=tri(1), 1=saw(3), 2=sine(0), 3=square(2), 4=pulse(4).
    // ISA 16x4 f32 A layout: lane L -> M = L%16 ; VGPR0 -> K = 2*(L>=16) ; VGPR1 -> K+1.
    const int kperm[8] = {1, 3, 0, 2, 4, -1, -1, -1};
    const int k0 = (int)half * 2;
    auto wv = [&](int r, int k) -> float {
        const int c = (k >= 0 && k < 8) ? kperm[k] : -1;
        return (r < 5 && c >= 0 && c < 5 && r != c) ? W[r * 5 + c] : 0.0f;
    };
    v2f afrag0, afrag1;
    afrag0.x = wv((int)m, k0);
    afrag0.y = wv((int)m, k0 + 1);
    afrag1.x = wv((int)m, k0 + 4);   // only (half==0, K=4 -> pulse) nonzero
    afrag1.y = wv((int)m, k0 + 5);

    const unsigned waveId = (blockIdx.x * blockDim.x + threadIdx.x) >> 5;
    const unsigned nWaves = (gridDim.x * blockDim.x) >> 5;

    for (unsigned tile = waveId; tile < numTiles; tile += nWaves) {
        const int  s_idx = (int)(tile * 16u + m);
        const bool valid = s_idx < n;
        const float t = valid ? (float)s_idx * inv_sr : 0.0f;

        // --- B fragments (amp-scaled oscillator values, permuted K order).
        // B 4x16 layout mirrors A: lane L -> N = L%16 ; VGPR0 -> K = 2*(L>=16).
        v2f b0, b1;
        if (half == 0) {
            // K=0: triangle (sin+asin), K=1: sawtooth (0 trans), K=4: pulse (1 sin)
            const float tri  = a[1] * TWO_OVER_PI_F * asinf(sin_fast(om[1] * t));
            const float sw   = a[3] * TWO_OVER_PI_F * (om[3] * t - PI_F);
            const float puls = a[4] * ((sin_fast(om[4] * t) < pw) ? 1.0f : -1.0f);
            b0.x = tri;  b0.y = sw;
            b1.x = puls; b1.y = 0.0f;
        } else {
            // K=2: sine (1 sin), K=3: square (1 sin)
            const float sine = a[0] * sin_fast(om[0] * t);
            const float s2   = sin_fast(om[2] * t);
            const float sq   = a[2] * ((s2 > 0.0f) ? 1.0f : ((s2 < 0.0f) ? -1.0f : 0.0f));
            b0.x = sine; b0.y = sq;
            b1.x = 0.0f; b1.y = 0.0f;
        }
        if (!valid) { b0.x = 0.0f; b0.y = 0.0f; b1.x = 0.0f; b1.y = 0.0f; }

        // Control flow rejoined: EXEC all-1s for the matrix ops.
        v8f acc = {0.f, 0.f, 0.f, 0.f, 0.f, 0.f, 0.f, 0.f};
        acc = __builtin_amdgcn_wmma_f32_16x16x4_f32(
                  false, afrag0, false, b0, (short)0, acc, false, false);
        acc = __builtin_amdgcn_wmma_f32_16x16x4_f32(
                  false, afrag1, false, b1, (short)0, acc, false, false);

        // D: lane L (<16), VGPR r -> mod[carrier r][sample L], r=0..4.
        // Ship carriers 3,4 to the upper half so the 5 epilogue sins run 3/2.
        const float md3 = __shfl(acc[3], (int)m, 32);
        const float md4 = __shfl(acc[4], (int)m, 32);

        float partial;
        if (half == 0) {
            partial  = a[0] * sin_fast(cc[0] + TWO_PI_F * acc[0]);
            partial += a[1] * sin_fast(cc[1] + TWO_PI_F * acc[1]);
            partial += a[2] * sin_fast(cc[2] + TWO_PI_F * acc[2]);
        } else {
            partial  = a[3] * sin_fast(cc[3] + TWO_PI_F * md3);
            partial += a[4] * sin_fast(cc[4] + TWO_PI_F * md4);
        }
        // Combine upper-half partial into lower half; lanes 0..15 store.
        const float other = __shfl(partial, (int)(lane | 16u), 32);
        if (half == 0 && valid) {
            out[s_idx] = partial + other;   // 64B coalesced store per wave-tile
        }
    }
}

extern "C" void kernel_launch(void* const* d_in, const int* in_sizes, int n_in,
                              void* d_out, int out_size, void* d_ws, size_t ws_size,
                              hipStream_t stream) {
    (void)in_sizes; (void)n_in; (void)d_ws; (void)ws_size; (void)out_size;
    const float* freqs       = (const float*)d_in[0];
    const float* amps        = (const float*)d_in[1];
    const float* pulse_width = (const float*)d_in[2];
    const float* W           = (const float*)d_in[3];
    const int*   duration    = (const int*)d_in[4];
    const int*   sample_rate = (const int*)d_in[5];
    float*       out         = (float*)d_out;

    // n = 5,760,000 -> 360,000 tiles of 16 samples; 16384 waves grid-stride.
    dim3 block(256);
    dim3 grid(2048);
    fm_synth_wmma<<<grid, block, 0, stream>>>(freqs, amps, pulse_width, W,
                                              duration, sample_rate, out);
}